// MDGRU_40638980555208
// MI455X (gfx1250) — compile-verified
//
#include <hip/hip_runtime.h>

typedef float v2f __attribute__((ext_vector_type(2)));
typedef float v4f __attribute__((ext_vector_type(4)));
typedef float v8f __attribute__((ext_vector_type(8)));

#define TSEQ 2048
#define BB   64
#define DK   256
#define UC   256
#define C3   768

// full-precision f32 WMMA: D(16x16) = A(16x4) * B(4x16) + C
static __device__ inline v8f wmma4(v2f a, v2f b, v8f c) {
    return __builtin_amdgcn_wmma_f32_16x16x4_f32(
        /*neg_a=*/false, a, /*neg_b=*/false, b,
        /*c_mod=*/(short)0, c, /*reuse_a=*/false, /*reuse_b=*/false);
}

// map state element (b,u) -> flat float index in A-fragment order:
// frag[m][j][lane]*4+c, lane = hi*16 + (b&15),
// float4 = { A(row, 8j+2hi), A(row, 8j+2hi+1), A(row, 8j+4+2hi), A(row, 8j+4+2hi+1) }
static __device__ inline int fragIdx(int b, int u) {
    const int m  = b >> 4, lb = b & 15;
    const int j  = u >> 3, r  = u & 7;
    const int hi = (r >> 1) & 1;
    const int c  = (r & 1) | ((r >> 2) << 1);
    return (((m * 32 + j) * 32) + hi * 16 + lb) * 4 + c;
}

// -------------------------------------------------------------------------
// Kernel 0: pack recurrent weights into per-lane WMMA B-fragment order.
// Wpack[tt][j][lane] = float4{ B(k0,col), B(k0+1,col), B(k0+4,col), B(k0+5,col) }
//   tt: 16-col tile (0..47), col = tt*16+(lane&15), k0 = 8j + 2*(lane>>4)
// -------------------------------------------------------------------------
__global__ __launch_bounds__(256) void mdgru_pack_kernel(
    const float* __restrict__ Urec, float4* __restrict__ Wp)
{
    const int id   = blockIdx.x * 256 + threadIdx.x;   // [0, 48*32*32)
    const int lane = id & 31;
    const int j    = (id >> 5) & 31;
    const int tt   = id >> 10;
    const int l15  = lane & 15, hi = lane >> 4;
    const int col  = tt * 16 + l15;
    const int k0   = 8 * j + 2 * hi;
    float4 f;
    f.x = Urec[(size_t)(k0    ) * C3 + col];
    f.y = Urec[(size_t)(k0 + 1) * C3 + col];
    f.z = Urec[(size_t)(k0 + 4) * C3 + col];
    f.w = Urec[(size_t)(k0 + 5) * C3 + col];
    Wp[id] = f;
}

// -------------------------------------------------------------------------
// Kernel 1: proj[t][b][c] = X[b][t0+t][:] @ kernel[:,c] + bias[c]
// grid = (Cchunk, 6); block = 256 threads (8 waves).
// -------------------------------------------------------------------------
__global__ __launch_bounds__(256) void mdgru_proj_kernel(
    const float* __restrict__ X, const float* __restrict__ W,
    const float* __restrict__ bias, float* __restrict__ P, int t0)
{
    __shared__ float As[64 * 18];
    __shared__ float Bs[16 * 132];

    const int tb   = blockIdx.x;
    const int cblk = blockIdx.y;
    const int tid  = threadIdx.x;
    const int lane = tid & 31;
    const int w    = tid >> 5;
    const int l15  = lane & 15;
    const int hi   = lane >> 4;
    const int wm   = w & 1;
    const int wn   = w >> 1;
    const int t    = t0 + tb;

    v8f acc[2][2];
    for (int i = 0; i < 2; ++i)
        for (int j = 0; j < 2; ++j)
            acc[i][j] = (v8f){0.f,0.f,0.f,0.f,0.f,0.f,0.f,0.f};

    for (int k0 = 0; k0 < DK; k0 += 16) {
        {
            const int r  = tid >> 2;
            const int kk = (tid & 3) * 4;
            const float* src = X + ((size_t)r * TSEQ + t) * DK + k0 + kk;
            #pragma unroll
            for (int j = 0; j < 4; ++j) As[r * 18 + kk + j] = src[j];
        }
        {
            const int r  = tid >> 4;
            const int c0 = (tid & 15) * 8;
            const float* src = W + (size_t)(k0 + r) * C3 + cblk * 128 + c0;
            #pragma unroll
            for (int j = 0; j < 8; ++j) Bs[r * 132 + c0 + j] = src[j];
        }
        __syncthreads();
        #pragma unroll
        for (int kk = 0; kk < 16; kk += 4) {
            v2f af[2], bf[2];
            #pragma unroll
            for (int mt = 0; mt < 2; ++mt) {
                const int mr = wm * 32 + mt * 16 + l15;
                af[mt].x = As[mr * 18 + kk + 2 * hi];
                af[mt].y = As[mr * 18 + kk + 2 * hi + 1];
            }
            #pragma unroll
            for (int nt = 0; nt < 2; ++nt) {
                const int cn = wn * 32 + nt * 16 + l15;
                bf[nt].x = Bs[(kk + 2 * hi)     * 132 + cn];
                bf[nt].y = Bs[(kk + 2 * hi + 1) * 132 + cn];
            }
            #pragma unroll
            for (int mt = 0; mt < 2; ++mt)
                #pragma unroll
                for (int nt = 0; nt < 2; ++nt)
                    acc[mt][nt] = wmma4(af[mt], bf[nt], acc[mt][nt]);
        }
        __syncthreads();
    }

    #pragma unroll
    for (int mt = 0; mt < 2; ++mt) {
        #pragma unroll
        for (int nt = 0; nt < 2; ++nt) {
            const int col = cblk * 128 + wn * 32 + nt * 16 + l15;
            const float bv = bias[col];
            #pragma unroll
            for (int q = 0; q < 8; ++q) {
                const int row = wm * 32 + mt * 16 + q + 8 * hi;
                P[((size_t)tb * BB + row) * C3 + col] = acc[mt][nt][q] + bv;
            }
        }
    }
}

// -------------------------------------------------------------------------
// Kernel 2: sequential GRU scan. One persistent workgroup, 1024 threads
// (32 waves). LDS: a-state fragments (64KB) + z/r/h scratch S (129KB) +
// (r*a) fragments (64KB) = 257KB of the 320KB WGP LDS.
// Per j in the GEMM loops: 1 coalesced global b128 (packed weights, L2) +
// b128 fragment loads from LDS + WMMAs. P[t+1] is prefetched each step to
// hide HBM latency on the serial path.
// -------------------------------------------------------------------------
#define S_STRIDE 516
#define FRAG_FLOATS (4 * 32 * 32 * 4)     // 16384 floats = 64KB

__global__ __launch_bounds__(1024) void mdgru_scan_kernel(
    const v4f* __restrict__ Wpack, const float* __restrict__ P,
    float* __restrict__ aState, float* __restrict__ out,
    int Csteps, int initA, int lastChunk)
{
    extern __shared__ float smem[];
    float* aFragF  = smem;                                   // 16384 floats
    float* S       = smem + FRAG_FLOATS;                     // 64 x 516
    float* raFragF = smem + FRAG_FLOATS + BB * S_STRIDE;     // 16384 floats
    v4f*   aF      = (v4f*)aFragF;
    v4f*   raF     = (v4f*)raFragF;

    const int tid  = threadIdx.x;
    const int lane = tid & 31;
    const int w    = tid >> 5;           // wave 0..31
    const int l15  = lane & 15;
    const int hi   = lane >> 4;

    for (int e = tid; e < BB * UC; e += 1024) {
        const int b = e >> 8, u = e & 255;
        aFragF[fragIdx(b, u)] = initA ? 0.0f : aState[e];
    }
    __syncthreads();

    for (int t = 0; t < Csteps; ++t) {
        const float* Pt = P + (size_t)t * BB * C3;

        // pull next step's 192KB projection slab toward the caches while
        // the GEMMs run (1024 threads x 2 lines tile the whole slab)
        {
            const int tn = (t + 1 < Csteps) ? (t + 1) : t;
            const char* pn = (const char*)(P + (size_t)tn * BB * C3)
                           + (size_t)tid * 192;
            __builtin_prefetch(pn, 0, 0);
            __builtin_prefetch(pn + 128, 0, 0);
        }

        // ---- Phase A: S[:,0:512) = a @ [Uz|Ur] --------------------------
        {
            const v4f* Wp = Wpack + (size_t)w * (32 * 32);
            v8f acc[4];
            #pragma unroll
            for (int m = 0; m < 4; ++m)
                acc[m] = (v8f){0.f,0.f,0.f,0.f,0.f,0.f,0.f,0.f};
            for (int j = 0; j < 32; ++j) {
                const v4f bw = Wp[j * 32 + lane];
                const v2f b0 = __builtin_shufflevector(bw, bw, 0, 1);
                const v2f b1 = __builtin_shufflevector(bw, bw, 2, 3);
                #pragma unroll
                for (int m = 0; m < 4; ++m) {
                    const v4f av = aF[(m * 32 + j) * 32 + lane];
                    acc[m] = wmma4(__builtin_shufflevector(av, av, 0, 1), b0, acc[m]);
                    acc[m] = wmma4(__builtin_shufflevector(av, av, 2, 3), b1, acc[m]);
                }
            }
            const int col = w * 16 + l15;
            #pragma unroll
            for (int m = 0; m < 4; ++m)
                #pragma unroll
                for (int q = 0; q < 8; ++q)
                    S[(m * 16 + q + 8 * hi) * S_STRIDE + col] = acc[m][q];
        }
        __syncthreads();

        // ---- Phase B: z = sigmoid(Sz+tz); r = sigmoid(Sr+tr); ra = r*a --
        for (int e = tid; e < BB * UC; e += 1024) {
            const int b = e >> 8, u = e & 255;
            const float sz = S[b * S_STRIDE + u]       + Pt[b * C3 + u];
            const float sr = S[b * S_STRIDE + 256 + u] + Pt[b * C3 + 256 + u];
            const float z = 1.0f / (1.0f + __expf(-sz));
            const float r = 1.0f / (1.0f + __expf(-sr));
            const int fi = fragIdx(b, u);
            raFragF[fi] = r * aFragF[fi];
            S[b * S_STRIDE + u] = z;
        }
        __syncthreads();

        // ---- Phase C: S[:,256:512) = (r*a) @ Uh -------------------------
        {
            const int tt = 32 + (w >> 1);        // Uh column tile
            const int mb = (w & 1) * 2;          // this wave's M pair
            const v4f* Wp = Wpack + (size_t)tt * (32 * 32);
            v8f acc[2];
            acc[0] = (v8f){0.f,0.f,0.f,0.f,0.f,0.f,0.f,0.f};
            acc[1] = (v8f){0.f,0.f,0.f,0.f,0.f,0.f,0.f,0.f};
            for (int j = 0; j < 32; ++j) {
                const v4f bw = Wp[j * 32 + lane];
                const v2f b0 = __builtin_shufflevector(bw, bw, 0, 1);
                const v2f b1 = __builtin_shufflevector(bw, bw, 2, 3);
                #pragma unroll
                for (int mi = 0; mi < 2; ++mi) {
                    const v4f av = raF[((mb + mi) * 32 + j) * 32 + lane];
                    acc[mi] = wmma4(__builtin_shufflevector(av, av, 0, 1), b0, acc[mi]);
                    acc[mi] = wmma4(__builtin_shufflevector(av, av, 2, 3), b1, acc[mi]);
                }
            }
            const int col = (w >> 1) * 16 + l15;
            #pragma unroll
            for (int mi = 0; mi < 2; ++mi)
                #pragma unroll
                for (int q = 0; q < 8; ++q)
                    S[((mb + mi) * 16 + q + 8 * hi) * S_STRIDE + 256 + col] = acc[mi][q];
        }
        __syncthreads();

        // ---- Phase D: a = z*a + (1-z)*tanh(Sh + th) ---------------------
        for (int e = tid; e < BB * UC; e += 1024) {
            const int b = e >> 8, u = e & 255;
            const float z = S[b * S_STRIDE + u];
            const float h = tanhf(S[b * S_STRIDE + 256 + u] + Pt[b * C3 + 512 + u]);
            const int fi = fragIdx(b, u);
            const float a0 = aFragF[fi];
            aFragF[fi] = z * a0 + (1.0f - z) * h;
        }
        __syncthreads();
    }

    for (int e = tid; e < BB * UC; e += 1024) {
        const int b = e >> 8, u = e & 255;
        const float v = aFragF[fragIdx(b, u)];
        aState[e] = v;
        if (lastChunk) out[e] = v;
    }
}

// -------------------------------------------------------------------------
extern "C" void kernel_launch(void* const* d_in, const int* in_sizes, int n_in,
                              void* d_out, int out_size, void* d_ws, size_t ws_size,
                              hipStream_t stream) {
    const float* X    = (const float*)d_in[0];   // (64, 2048, 256)
    const float* W    = (const float*)d_in[1];   // (256, 768)
    const float* Urec = (const float*)d_in[2];   // (256, 768)
    const float* bias = (const float*)d_in[3];   // (768,)
    float* out = (float*)d_out;                  // (64, 256)

    // workspace layout: [ a state | packed weights | proj chunk buffer ]
    float*  aState = (float*)d_ws;
    const size_t stateBytes = (size_t)BB * UC * sizeof(float);         // 64 KB
    const size_t packBytes  = (size_t)48 * 32 * 32 * sizeof(float4);   // 768 KB
    float4* Wpack = (float4*)((char*)d_ws + stateBytes);
    float*  P     = (float*)((char*)d_ws + stateBytes + packBytes);

    const size_t perT = (size_t)BB * C3 * sizeof(float);
    int C = TSEQ;
    while (C > 16 && stateBytes + packBytes + (size_t)C * perT > ws_size) C >>= 1;

    const int chunks = TSEQ / C;
    const size_t ldsBytes =
        (size_t)(FRAG_FLOATS * 2 + BB * S_STRIDE) * sizeof(float);     // 257 KB

    mdgru_pack_kernel<<<192, 256, 0, stream>>>(Urec, Wpack);
    for (int ci = 0; ci < chunks; ++ci) {
        const int t0 = ci * C;
        mdgru_proj_kernel<<<dim3(C, 6), 256, 0, stream>>>(X, W, bias, P, t0);
        mdgru_scan_kernel<<<1, 1024, ldsBytes, stream>>>(
            (const v4f*)Wpack, P, aState, out, C,
            ci == 0 ? 1 : 0, ci == chunks - 1 ? 1 : 0);
    }
}